// Subsampling_Layer_3D_14482629722299
// MI455X (gfx1250) — compile-verified
//
#include <hip/hip_runtime.h>
#include <hip/hip_bf16.h>

// Problem constants (from reference): B=8, D=64, H=128, W=128, 512*500 traj pts.
#define B_   8
#define D_   64
#define H_   128
#define W_   128
#define HW_  (H_ * W_)
#define DHW_ (D_ * H_ * W_)
#define NPTS_TOTAL_ (512 * 500)

typedef __attribute__((ext_vector_type(2))) float v2f;
typedef __attribute__((ext_vector_type(8))) float v8f;

// ---- f32 WMMA 16x16x4: D = A(16x4) * B(4x16) + C --------------------------
// f32 WMMA has no A/B negate (NEG = CNeg only) -> negate in the fragment.
__device__ __forceinline__ v8f wmma4(v2f a, v2f b, v8f c) {
  return __builtin_amdgcn_wmma_f32_16x16x4_f32(false, a, false, b, (short)0, c,
                                               false, false);
}

// Centered-orthonormal inverse-DFT matrix element (symmetric):
//   M[i][j] = (1/sqrt(N)) * exp(+2*pi*1i * (i - N/2)(j - N/2) / N)
__device__ __forceinline__ void dft_elem(int i, int j, int n, float scale,
                                         float& cr, float& ci) {
  int c = n >> 1;
  int t = ((i - c) * (j - c)) % n;
  t += (t >> 31) & n;                      // positive remainder
  float ph = (float)(2 * t) / (float)n;    // in [0, 2)
  float s, co;
  sincospif(ph, &s, &co);
  cr = co * scale;
  ci = s * scale;
}

// Pack twiddles in WMMA-fragment order:
//   mp[kp * n + x] = { M[2kp][x].re, .im, M[2kp+1][x].re, .im }
// One b128 load per lane then yields a full 2-K complex fragment.
__global__ void __launch_bounds__(256)
build_twiddles(float4* __restrict__ mp, int n, float scale) {
  int idx = blockIdx.x * 256 + threadIdx.x;
  int pairs = (n >> 1) * n;
  if (idx >= pairs) return;
  int x = idx % n;
  int kp = idx / n;
  float cr0, ci0, cr1, ci1;
  dft_elem(2 * kp, x, n, scale, cr0, ci0);
  dft_elem(2 * kp + 1, x, n, scale, cr1, ci1);
  mp[idx] = make_float4(cr0, ci0, cr1, ci1);
}

// ---- Fused trilinear degrid + adjoint regrid (gather -> scatter atomics) ---
__global__ void __launch_bounds__(256)
degrid_regrid(const float* __restrict__ gin,   // (B, DHW, 2)
              const float* __restrict__ traj,  // (NPTS_TOTAL, 3)
              float* __restrict__ gout)        // (B, DHW, 2), pre-zeroed
{
  int tid = blockIdx.x * 256 + threadIdx.x;
  const int total = B_ * NPTS_TOTAL_;
  if (tid >= total) return;
  int b = tid / NPTS_TOTAL_;
  int n = tid - b * NPTS_TOTAL_;

  float p0 = traj[n * 3 + 0];
  float p1 = traj[n * 3 + 1];
  float p2 = traj[n * 3 + 2];
  p0 = fminf(fmaxf(p0, -32.f), 32.f);
  p1 = fminf(fmaxf(p1, -64.f), 64.f);
  p2 = fminf(fmaxf(p2, -64.f), 64.f);
  float idx0 = p0 + 32.f, idx1 = p1 + 64.f, idx2 = p2 + 64.f;
  float i0f = fminf(fmaxf(floorf(idx0), 0.f), (float)(D_ - 2));
  float i1f = fminf(fmaxf(floorf(idx1), 0.f), (float)(H_ - 2));
  float i2f = fminf(fmaxf(floorf(idx2), 0.f), (float)(W_ - 2));
  float f0 = fminf(fmaxf(idx0 - i0f, 0.f), 1.f);
  float f1 = fminf(fmaxf(idx1 - i1f, 0.f), 1.f);
  float f2 = fminf(fmaxf(idx2 - i2f, 0.f), 1.f);
  int i0 = (int)i0f, i1 = (int)i1f, i2 = (int)i2f;

  const float* g = gin + (size_t)b * DHW_ * 2;
  int   fl[8];
  float ww[8];
  float kr = 0.f, ki = 0.f;
#pragma unroll
  for (int c = 0; c < 8; ++c) {
    int a = (c >> 2) & 1, bb = (c >> 1) & 1, cc = c & 1;
    float w = (a ? f0 : 1.f - f0) * (bb ? f1 : 1.f - f1) * (cc ? f2 : 1.f - f2);
    int fi = (i0 + a) * HW_ + (i1 + bb) * W_ + (i2 + cc);
    fl[c] = fi;
    ww[c] = w;
    kr += w * g[(size_t)fi * 2];
    ki += w * g[(size_t)fi * 2 + 1];
  }
  float* go = gout + (size_t)b * DHW_ * 2;
#pragma unroll
  for (int c = 0; c < 8; ++c) {
    atomicAdd(&go[(size_t)fl[c] * 2], ww[c] * kr);
    atomicAdd(&go[(size_t)fl[c] * 2 + 1], ww[c] * ki);
  }
}

// ---- Transform along the CONTIGUOUS axis: out[r,x] = sum_k in[r,k] M[k,x]
// A = data rows, B = packed twiddles.  Each wave computes TWO 16x16 tiles
// (x0, x0+16) sharing the A fragment: 8 WMMA per 3 b128-class loads.
template <int N>
__global__ void __launch_bounds__(32)
dft_last(const float2* __restrict__ in, float2* __restrict__ out,
         const float4* __restrict__ mp) {
  const int xpairs = N / 32;
  int lane = threadIdx.x;
  int p = lane & 15, hi = lane >> 4;
  int tile = blockIdx.x;
  int txp = tile % xpairs;
  int tr = tile / xpairs;
  int r0 = tr * 16, x0 = txp * 32;

  v8f accr0 = {}, acci0 = {}, accr1 = {}, acci1 = {};
  const float4* arow4 = (const float4*)(in + (size_t)(r0 + p) * N);
  const float4* mrow = mp + x0 + p;
#pragma unroll 4
  for (int k0 = 0; k0 < N; k0 += 4) {
    int kh = (k0 >> 1) + hi;  // k/2 for this lane's K pair
    float4 d = arow4[kh];
    v2f ar  = {d.x, d.z};
    v2f ai  = {d.y, d.w};
    v2f ain = {-d.y, -d.w};
    float4 m0 = mrow[(size_t)kh * N];
    float4 m1 = mrow[(size_t)kh * N + 16];
    v2f br0 = {m0.x, m0.z}, bi0 = {m0.y, m0.w};
    v2f br1 = {m1.x, m1.z}, bi1 = {m1.y, m1.w};
    accr0 = wmma4(ar, br0, accr0);
    accr0 = wmma4(ain, bi0, accr0);
    acci0 = wmma4(ar, bi0, acci0);
    acci0 = wmma4(ai, br0, acci0);
    accr1 = wmma4(ar, br1, accr1);
    accr1 = wmma4(ain, bi1, accr1);
    acci1 = wmma4(ar, bi1, acci1);
    acci1 = wmma4(ai, br1, acci1);
  }
#pragma unroll
  for (int v = 0; v < 8; ++v) {
    int row = r0 + 8 * hi + v;
    out[(size_t)row * N + x0 + p]      = make_float2(accr0[v], acci0[v]);
    out[(size_t)row * N + x0 + 16 + p] = make_float2(accr1[v], acci1[v]);
  }
}

// ---- Transform along a STRIDED axis: out[s,x,n] = sum_k M[x,k] in[s,k,n]
// A = packed twiddles (symmetric M), B = data.  Two n-tiles per wave.
template <int L>
__global__ void __launch_bounds__(32)
dft_mid(const float2* __restrict__ in, float2* __restrict__ out,
        const float4* __restrict__ mp, int S, int npairs) {
  const int tiles_x = L / 16;
  int lane = threadIdx.x;
  int p = lane & 15, hi = lane >> 4;
  int t = blockIdx.x;
  int tnp = t % npairs;
  t /= npairs;
  int tx = t % tiles_x;
  int slice = t / tiles_x;
  size_t base = (size_t)slice * L * S;
  int x0 = tx * 16, n0 = tnp * 32;

  v8f accr0 = {}, acci0 = {}, accr1 = {}, acci1 = {};
  const float4* mrow = mp + x0 + p;
#pragma unroll 4
  for (int k0 = 0; k0 < L; k0 += 4) {
    int kh = (k0 >> 1) + hi;
    int k = k0 + 2 * hi;
    float4 m = mrow[(size_t)kh * L];  // M[x0+p][k..k+1] via symmetry
    v2f ar  = {m.x, m.z};
    v2f ai  = {m.y, m.w};
    v2f ain = {-m.y, -m.w};
    float2 d00 = in[base + (size_t)k * S + n0 + p];
    float2 d01 = in[base + (size_t)(k + 1) * S + n0 + p];
    float2 d10 = in[base + (size_t)k * S + n0 + 16 + p];
    float2 d11 = in[base + (size_t)(k + 1) * S + n0 + 16 + p];
    v2f br0 = {d00.x, d01.x}, bi0 = {d00.y, d01.y};
    v2f br1 = {d10.x, d11.x}, bi1 = {d10.y, d11.y};
    accr0 = wmma4(ar, br0, accr0);
    accr0 = wmma4(ain, bi0, accr0);
    acci0 = wmma4(ar, bi0, acci0);
    acci0 = wmma4(ai, br0, acci0);
    accr1 = wmma4(ar, br1, accr1);
    accr1 = wmma4(ain, bi1, accr1);
    acci1 = wmma4(ar, bi1, acci1);
    acci1 = wmma4(ai, br1, acci1);
  }
#pragma unroll
  for (int v = 0; v < 8; ++v) {
    int x = x0 + 8 * hi + v;
    out[base + (size_t)x * S + n0 + p]      = make_float2(accr0[v], acci0[v]);
    out[base + (size_t)x * S + n0 + 16 + p] = make_float2(accr1[v], acci1[v]);
  }
}

extern "C" void kernel_launch(void* const* d_in, const int* in_sizes, int n_in,
                              void* d_out, int out_size, void* d_ws,
                              size_t ws_size, hipStream_t stream) {
  const float* grid_in = (const float*)d_in[0];  // (B,1,1,D,H,W,2) fp32
  const float* traj    = (const float*)d_in[1];  // (512,500,3) fp32
  float* out = (float*)d_out;                    // (B,1,D,H,W,2) fp32

  // Workspace layout: [G: B*DHW complex = 64 MiB][M128 pack: 128 KiB][M64 pack: 32 KiB]
  float*  G    = (float*)d_ws;
  const size_t gbytes = (size_t)B_ * DHW_ * 2 * sizeof(float);
  float4* M128 = (float4*)((char*)d_ws + gbytes);
  float4* M64  = M128 + (W_ / 2) * W_;

  hipMemsetAsync(G, 0, gbytes, stream);

  // 0) twiddle tables (tiny; L2-resident for the whole run)
  {
    int pairs128 = (W_ / 2) * W_;  // 8192
    int pairs64  = (D_ / 2) * D_;  // 2048
    build_twiddles<<<(pairs128 + 255) / 256, 256, 0, stream>>>(
        M128, W_, 0.08838834764831845f);  // 1/sqrt(128)
    build_twiddles<<<(pairs64 + 255) / 256, 256, 0, stream>>>(
        M64, D_, 0.125f);                 // 1/sqrt(64)
  }

  // 1) fused degrid + adjoint regrid into G
  {
    int total = B_ * NPTS_TOTAL_;
    degrid_regrid<<<(total + 255) / 256, 256, 0, stream>>>(grid_in, traj, G);
  }

  // 2) W axis (contiguous): G -> out.  rows = B*D*H = 65536, two x-tiles/wave
  {
    int rows = B_ * D_ * H_;
    dft_last<W_><<<(rows / 16) * (W_ / 32), 32, 0, stream>>>(
        (const float2*)G, (float2*)out, M128);
  }
  // 3) H axis (stride W): out -> G.  slices = B*D = 512
  {
    int npairs = W_ / 32;  // 4
    dft_mid<H_><<<(B_ * D_) * (H_ / 16) * npairs, 32, 0, stream>>>(
        (const float2*)out, (float2*)G, M128, W_, npairs);
  }
  // 4) D axis (stride H*W): G -> out.  slices = B = 8
  {
    int npairs = HW_ / 32;  // 512
    dft_mid<D_><<<B_ * (D_ / 16) * npairs, 32, 0, stream>>>(
        (const float2*)G, (float2*)out, M64, HW_, npairs);
  }
}